// GraphSageDQN_15307263443567
// MI455X (gfx1250) — compile-verified
//
#include <hip/hip_runtime.h>
#include <hip/hip_bf16.h>

// GraphSAGE-DQN forward for MI455X (gfx1250, wave32).
//   conv1: h1 = mean_agg(x)@W1_l + b1 + x@W1_r
//   conv2: h2 = mean_agg(h1)@W2_l + b2 + h1@W2_r
//   q = relu(mean(h2,axis=0)@Wf1 + bf1)@Wf2 + bf2
// GEMMs use V_WMMA_F32_16X16X4_F32 (exact f32 semantics, matches reference).

#define NF 128   // node feature dim / embedding dim (K per matrix)
#define ND 128   // output embedding dim
#define NH 256   // hidden dim of head
#define NA 32    // actions

typedef __attribute__((ext_vector_type(2))) float v2f;
typedef __attribute__((ext_vector_type(8))) float v8f;

// ---------------------------------------------------------------- scatter ---
// 32 lanes per edge; each lane handles a 4-float chunk of the 128-float row.
__global__ void k_scatter(const float* __restrict__ feat,
                          const int* __restrict__ eidx,
                          float* __restrict__ agg,
                          float* __restrict__ deg,
                          int E, int do_deg) {
  long long tid = (long long)blockIdx.x * blockDim.x + threadIdx.x;
  long long e = tid >> 5;
  if (e >= E) return;
  int c = ((int)tid & 31) << 2;           // 0..124 step 4
  int src = eidx[e];
  int dst = eidx[(long long)E + e];
  const float4 v = *(const float4*)(feat + (size_t)src * NF + c);
  float* base = agg + (size_t)dst * NF + c;
  atomicAdd(base + 0, v.x);
  atomicAdd(base + 1, v.y);
  atomicAdd(base + 2, v.z);
  atomicAdd(base + 3, v.w);
  if (do_deg && ((int)tid & 31) == 0) atomicAdd(&deg[dst], 1.0f);
}

// ----------------------------------------------------------- divide by deg ---
__global__ void k_divdeg(float* __restrict__ agg, const float* __restrict__ deg,
                         int total) {
  int tid = blockIdx.x * blockDim.x + threadIdx.x;
  if (tid >= total) return;
  int row = tid >> 7;                     // NF == 128
  agg[tid] = agg[tid] / fmaxf(deg[row], 1.0f);
}

// ------------------------------------------------------------- WMMA GEMM ----
// out[n_rows, ND] = A0@B0 + A1@B1 + bias,  A* are [n_rows,128], B* are [128,ND].
// One wave32 computes one 16x16 output tile via V_WMMA_F32_16X16X4_F32.
//
// Fragment layouts (CDNA5 ISA 7.12.2, 32-bit):
//   A 16x4 : lane L: M = L&15; VGPR0 = A[M][2*(L>>4)],  VGPR1 = A[M][2*(L>>4)+1]
//   B 4x16 : lane L: N = L&15; VGPR0 = B[2*(L>>4)][N],  VGPR1 = B[2*(L>>4)+1][N]
//   C/D    : lane L: N = L&15; VGPR i = D[i + 8*(L>>4)][N]
__device__ __forceinline__ v8f wmma_k128(v8f acc,
                                         const float* __restrict__ Arow,
                                         const float* __restrict__ Bcol) {
#pragma unroll 8
  for (int k = 0; k < NF; k += 4) {
    v2f a, b;
    a.x = Arow[k];
    a.y = Arow[k + 1];
    b.x = Bcol[(size_t)k * ND];
    b.y = Bcol[(size_t)(k + 1) * ND];
    acc = __builtin_amdgcn_wmma_f32_16x16x4_f32(
        /*neg_a=*/false, a, /*neg_b=*/false, b,
        /*c_mod=*/(short)0, acc, /*reuse_a=*/false, /*reuse_b=*/false);
  }
  return acc;
}

__global__ void __launch_bounds__(128)
k_sage_gemm(const float* __restrict__ A0, const float* __restrict__ B0,
            const float* __restrict__ A1, const float* __restrict__ B1,
            const float* __restrict__ bias, float* __restrict__ out,
            int n_rows) {
  const int lane = threadIdx.x & 31;
  const int wave = threadIdx.x >> 5;
  const int tilesN = ND / 16;                       // 8
  const int tilesM = (n_rows + 15) / 16;
  const int tile = blockIdx.x * 4 + wave;
  if (tile >= tilesM * tilesN) return;
  const int tM = tile / tilesN;
  const int tN = tile % tilesN;
  const int row0 = tM * 16, col0 = tN * 16;

  const int m    = lane & 15;
  const int half = lane >> 4;                       // 0 or 1 -> K offset 2*half
  const int rA   = min(row0 + m, n_rows - 1);       // clamp tail rows on load
  const int koff = half * 2;

  v8f acc = {};
  acc = wmma_k128(acc, A0 + (size_t)rA * NF + koff, B0 + (size_t)koff * ND + col0 + m);
  acc = wmma_k128(acc, A1 + (size_t)rA * NF + koff, B1 + (size_t)koff * ND + col0 + m);

  const float bz = bias[col0 + m];
#pragma unroll
  for (int i = 0; i < 8; ++i) {
    int r = row0 + i + 8 * half;
    if (r < n_rows) out[(size_t)r * ND + col0 + m] = acc[i] + bz;
  }
}

// -------------------------------------------------------------- column mean --
__global__ void k_colmean(const float* __restrict__ h, float* __restrict__ pooled,
                          int n_rows) {
  __shared__ float red[256];
  const int d = blockIdx.x;
  float s = 0.0f;
  for (int r = threadIdx.x; r < n_rows; r += blockDim.x)
    s += h[(size_t)r * ND + d];
  red[threadIdx.x] = s;
  __syncthreads();
  for (int off = 128; off > 0; off >>= 1) {
    if ((int)threadIdx.x < off) red[threadIdx.x] += red[threadIdx.x + off];
    __syncthreads();
  }
  if (threadIdx.x == 0) pooled[d] = red[0] / (float)n_rows;
}

// -------------------------------------------------------------------- head ---
__global__ void k_head(const float* __restrict__ pooled,
                       const float* __restrict__ Wf1, const float* __restrict__ bf1,
                       const float* __restrict__ Wf2, const float* __restrict__ bf2,
                       float* __restrict__ q) {
  __shared__ float pl[ND];
  __shared__ float hid[NH];
  const int t = threadIdx.x;
  if (t < ND) pl[t] = pooled[t];
  __syncthreads();
  float s = bf1[t];
#pragma unroll 8
  for (int k = 0; k < ND; ++k) s += pl[k] * Wf1[(size_t)k * NH + t];
  hid[t] = fmaxf(s, 0.0f);
  __syncthreads();
  if (t < NA) {
    float r = bf2[t];
#pragma unroll 8
    for (int k = 0; k < NH; ++k) r += hid[k] * Wf2[(size_t)k * NA + t];
    q[t] = r;
  }
}

// ------------------------------------------------------------------ launch ---
extern "C" void kernel_launch(void* const* d_in, const int* in_sizes, int n_in,
                              void* d_out, int out_size, void* d_ws, size_t ws_size,
                              hipStream_t stream) {
  const float* x    = (const float*)d_in[0];
  const int*   eidx = (const int*)d_in[1];
  const float* W1l  = (const float*)d_in[2];
  const float* b1   = (const float*)d_in[3];
  const float* W1r  = (const float*)d_in[4];
  const float* W2l  = (const float*)d_in[5];
  const float* b2   = (const float*)d_in[6];
  const float* W2r  = (const float*)d_in[7];
  const float* Wf1  = (const float*)d_in[8];
  const float* bf1  = (const float*)d_in[9];
  const float* Wf2  = (const float*)d_in[10];
  const float* bf2  = (const float*)d_in[11];

  const int N = in_sizes[0] / NF;     // 10000
  const int E = in_sizes[1] / 2;      // 640000
  const size_t NDsz = (size_t)N * NF;

  float* ws = (float*)d_ws;
  size_t o = 0;
  const size_t degpad = ((size_t)N + 255) & ~(size_t)255;
  float* deg    = ws + o; o += degpad;
  float* agg1   = ws + o; o += NDsz;
  float* agg2   = ws + o; o += NDsz;
  float* h1     = ws + o; o += NDsz;
  float* h2     = ws + o; o += NDsz;
  float* pooled = ws + o; o += ND;

  // zero the atomic accumulators (deg | agg1 | agg2 are contiguous)
  hipMemsetAsync(ws, 0, (degpad + 2 * NDsz) * sizeof(float), stream);

  const long long sthreads = (long long)E * 32;
  const int sblocks = (int)((sthreads + 255) / 256);
  const int dblocks = (int)((NDsz + 255) / 256);
  const int tiles   = ((N + 15) / 16) * (ND / 16);
  const int gblocks = (tiles + 3) / 4;

  // conv1
  k_scatter  <<<sblocks, 256, 0, stream>>>(x, eidx, agg1, deg, E, 1);
  k_divdeg   <<<dblocks, 256, 0, stream>>>(agg1, deg, (int)NDsz);
  k_sage_gemm<<<gblocks, 128, 0, stream>>>(agg1, W1l, x, W1r, b1, h1, N);
  // conv2 (degree reused: depends only on edge structure)
  k_scatter  <<<sblocks, 256, 0, stream>>>(h1, eidx, agg2, deg, E, 0);
  k_divdeg   <<<dblocks, 256, 0, stream>>>(agg2, deg, (int)NDsz);
  k_sage_gemm<<<gblocks, 128, 0, stream>>>(agg2, W2l, h1, W2r, b2, h2, N);
  // pool + head
  k_colmean  <<<ND, 256, 0, stream>>>(h2, pooled, N);
  k_head     <<<1, NH, 0, stream>>>(pooled, Wf1, bf1, Wf2, bf2, (float*)d_out);
}